// SSM_807453852218
// MI455X (gfx1250) — compile-verified
//
#include <hip/hip_runtime.h>
#include <math.h>

#define BATCH   16
#define SEQ     4096
#define D_INNER 192
#define D_STATE 4
#define DT_RANK 6
#define KDIM    192      // GEMM reduction dim = d_inner
#define NPAD    16       // x_dbl row padded 14 -> 16
#define NCHUNK  64
#define CHUNKLEN 64      // SEQ / NCHUNK
#define KSTEPS  (KDIM / 4)   // 48 WMMA k-steps per tile
#define TILES_PER_WAVE 4

typedef __attribute__((ext_vector_type(2))) float v2f;
typedef __attribute__((ext_vector_type(8))) float v8f;

// ---------------------------------------------------------------------------
// Phase 1: x_dbl[M=BATCH*SEQ][NPAD] = x[M][192] * W_xproj^T   (N = 14 used)
// V_WMMA_F32_16X16X4_F32, full f32 precision.
//   A 16x4:  VGPR j, lane half h -> K = k0 + j + 2h  (one float2 load)
//   B 4x16:  lane n = lane&15, VGPR j, half h -> K = k0 + j + 2h
//   C/D:     VGPR r -> M = r + 8*half, N = lane&15
// B operand (all 48 k-steps) is register-resident per wave and reused across
// TILES_PER_WAVE tiles; per tile all 48 A loads are issued before the WMMA
// chain so waits are pipelined instead of one round-trip per WMMA. Two
// accumulators break the C->D dependency chain.
// ---------------------------------------------------------------------------
__global__ __launch_bounds__(256) void xproj_wmma(const float* __restrict__ x,
                                                  const float* __restrict__ W,
                                                  float* __restrict__ xdbl) {
  const int lane = threadIdx.x & 31;
  const int wave = threadIdx.x >> 5;
  const int  m    = lane & 15;
  const int  half = lane >> 4;

  // ---- load entire B operand once (96 VGPRs), pad columns 14/15 to zero ----
  const int n = m;
  const float* brow = W + (long)((n < 14) ? n : 0) * KDIM + 2 * half;
  const float bscale = (n < 14) ? 1.0f : 0.0f;
  v2f b[KSTEPS];
#pragma unroll
  for (int i = 0; i < KSTEPS; ++i) {
    v2f t = *(const v2f*)(brow + 4 * i);
    t.x *= bscale; t.y *= bscale;
    b[i] = t;
  }

  const long tile0 = ((long)blockIdx.x * 8 + wave) * TILES_PER_WAVE;

  for (int tt = 0; tt < TILES_PER_WAVE; ++tt) {
    const long rowbase = (tile0 + tt) * 16;
    const float* arow = x + (rowbase + m) * KDIM + 2 * half;

    // ---- issue all A loads for this tile (independent, pipelined waits) ----
    v2f a[KSTEPS];
#pragma unroll
    for (int i = 0; i < KSTEPS; ++i) a[i] = *(const v2f*)(arow + 4 * i);

    // ---- 48 back-to-back WMMAs on two accumulator chains ----
    v8f acc0 = {}, acc1 = {};
#pragma unroll
    for (int i = 0; i < KSTEPS; i += 2) {
      acc0 = __builtin_amdgcn_wmma_f32_16x16x4_f32(false, a[i], false, b[i],
                                                   (short)0, acc0, false, false);
      acc1 = __builtin_amdgcn_wmma_f32_16x16x4_f32(false, a[i + 1], false, b[i + 1],
                                                   (short)0, acc1, false, false);
    }
    v8f acc = acc0 + acc1;

#pragma unroll
    for (int r = 0; r < 8; ++r) {
      long row = rowbase + r + 8 * half;
      xdbl[row * NPAD + m] = acc[r];
    }
  }
}

__device__ __forceinline__ float softplusf(float v) {
  return (v > 20.0f) ? v : log1pf(__expf(v));
}

// ---------------------------------------------------------------------------
// Phase 2a (WRITE_Y=false): local scan per chunk from h=0; store cumdecay + h_end.
// Phase 2c (WRITE_Y=true):  rescan with correct chunk-start state H0; emit y.
// grid = BATCH*NCHUNK blocks, 192 threads (one per d channel, 6 waves).
// ---------------------------------------------------------------------------
template <bool WRITE_Y>
__global__ __launch_bounds__(192) void scan_chunk(const float* __restrict__ x,
                                                  const float* __restrict__ xdbl,
                                                  const float* __restrict__ W_dt,
                                                  const float* __restrict__ b_dt,
                                                  const float* __restrict__ A_log,
                                                  const float* __restrict__ Dp,
                                                  const float* __restrict__ H0,
                                                  float* __restrict__ cp_out,
                                                  float* __restrict__ hl_out,
                                                  float* __restrict__ y_out) {
  const int b = blockIdx.x / NCHUNK;
  const int c = blockIdx.x % NCHUNK;
  const int d = threadIdx.x;

  float wdt[DT_RANK];
#pragma unroll
  for (int r = 0; r < DT_RANK; ++r) wdt[r] = W_dt[d * DT_RANK + r];
  const float bdt = b_dt[d];
  float Av[D_STATE];
#pragma unroll
  for (int n = 0; n < D_STATE; ++n) Av[n] = -__expf(A_log[d * D_STATE + n]);
  const float Dv = WRITE_Y ? Dp[d] : 0.0f;

  const long sidx = ((long)(b * NCHUNK + c) * D_INNER + d) * D_STATE;
  float h[D_STATE], cp[D_STATE];
#pragma unroll
  for (int n = 0; n < D_STATE; ++n) {
    h[n]  = WRITE_Y ? H0[sidx + n] : 0.0f;
    cp[n] = 1.0f;
  }

  const long l0 = (long)c * CHUNKLEN;
  for (int t = 0; t < CHUNKLEN; ++t) {
    const long row = (long)b * SEQ + l0 + t;
    const float* xr = xdbl + row * NPAD;        // block-uniform (broadcast)
    float dpre = bdt;
#pragma unroll
    for (int r = 0; r < DT_RANK; ++r) dpre += xr[r] * wdt[r];
    const float delta = softplusf(dpre);
    const float xv = x[row * D_INNER + d];
    const float dxu = delta * xv;
    if (WRITE_Y) {
      float y = Dv * xv;
#pragma unroll
      for (int n = 0; n < D_STATE; ++n) {
        float a = __expf(delta * Av[n]);
        h[n] = a * h[n] + dxu * xr[DT_RANK + n];              // B
        y += h[n] * xr[DT_RANK + D_STATE + n];                // C
      }
      y_out[row * D_INNER + d] = y;
    } else {
#pragma unroll
      for (int n = 0; n < D_STATE; ++n) {
        float a = __expf(delta * Av[n]);
        cp[n] *= a;
        h[n] = a * h[n] + dxu * xr[DT_RANK + n];
      }
    }
  }
  if (!WRITE_Y) {
#pragma unroll
    for (int n = 0; n < D_STATE; ++n) {
      cp_out[sidx + n] = cp[n];
      hl_out[sidx + n] = h[n];
    }
  }
}

// ---------------------------------------------------------------------------
// Phase 2b: serial prefix across chunks; one thread per (b,d,n) = 12288 threads.
// H0[chunk 0] = 0 ; H0[c+1] = cp[c]*H0[c] + hl[c]
// ---------------------------------------------------------------------------
__global__ __launch_bounds__(256) void chunk_prefix(const float* __restrict__ cp,
                                                    const float* __restrict__ hl,
                                                    float* __restrict__ H0) {
  const int idx = blockIdx.x * blockDim.x + threadIdx.x;
  if (idx >= BATCH * D_INNER * D_STATE) return;
  const int n = idx % D_STATE;
  const int d = (idx / D_STATE) % D_INNER;
  const int b = idx / (D_STATE * D_INNER);
  float H = 0.0f;
  for (int c = 0; c < NCHUNK; ++c) {
    const long s = ((long)(b * NCHUNK + c) * D_INNER + d) * D_STATE + n;
    H0[s] = H;
    H = cp[s] * H + hl[s];
  }
}

// ---------------------------------------------------------------------------
// Phase 3: in-place layernorm over d=192. One wave32 per (b,l) row,
// 6 elements/lane, butterfly reduce with __shfl_xor (wave32-correct).
// ---------------------------------------------------------------------------
__global__ __launch_bounds__(256) void layernorm_rows(float* __restrict__ y,
                                                      const float* __restrict__ gamma,
                                                      const float* __restrict__ beta) {
  const int lane = threadIdx.x & 31;
  const int wave = threadIdx.x >> 5;
  const long row = (long)blockIdx.x * 8 + wave;
  float* yr = y + row * D_INNER;
  float v[6];
  float s = 0.0f, sq = 0.0f;
#pragma unroll
  for (int j = 0; j < 6; ++j) {
    v[j] = yr[j * 32 + lane];
    s  += v[j];
    sq += v[j] * v[j];
  }
#pragma unroll
  for (int off = 16; off >= 1; off >>= 1) {
    s  += __shfl_xor(s, off, 32);
    sq += __shfl_xor(sq, off, 32);
  }
  const float mean = s * (1.0f / D_INNER);
  const float var  = sq * (1.0f / D_INNER) - mean * mean;
  const float rstd = rsqrtf(var + 1e-5f);
#pragma unroll
  for (int j = 0; j < 6; ++j) {
    const int col = j * 32 + lane;
    yr[col] = (v[j] - mean) * rstd * gamma[col] + beta[col];
  }
}

extern "C" void kernel_launch(void* const* d_in, const int* in_sizes, int n_in,
                              void* d_out, int out_size, void* d_ws, size_t ws_size,
                              hipStream_t stream) {
  const float* x     = (const float*)d_in[0];
  const float* Wxp   = (const float*)d_in[1];
  const float* Wdt   = (const float*)d_in[2];
  const float* bdt   = (const float*)d_in[3];
  const float* Alog  = (const float*)d_in[4];
  const float* Dp    = (const float*)d_in[5];
  const float* gamma = (const float*)d_in[6];
  const float* beta  = (const float*)d_in[7];
  float* y = (float*)d_out;

  float* ws   = (float*)d_ws;
  float* xdbl = ws;                                              // 1,048,576 f
  float* cp   = xdbl + (long)BATCH * SEQ * NPAD;                 //   786,432 f
  float* hl   = cp   + (long)BATCH * NCHUNK * D_INNER * D_STATE; //   786,432 f
  float* H0   = hl   + (long)BATCH * NCHUNK * D_INNER * D_STATE; //   786,432 f

  // 4096 M-tiles of 16 rows, 8 waves/block, 4 tiles/wave -> 128 blocks
  xproj_wmma<<<BATCH * SEQ / (16 * 8 * TILES_PER_WAVE), 256, 0, stream>>>(x, Wxp, xdbl);
  scan_chunk<false><<<BATCH * NCHUNK, 192, 0, stream>>>(x, xdbl, Wdt, bdt, Alog,
                                                        Dp, nullptr, cp, hl, y);
  chunk_prefix<<<(BATCH * D_INNER * D_STATE + 255) / 256, 256, 0, stream>>>(cp, hl, H0);
  scan_chunk<true><<<BATCH * NCHUNK, 192, 0, stream>>>(x, xdbl, Wdt, bdt, Alog,
                                                       Dp, H0, nullptr, nullptr, y);
  layernorm_rows<<<BATCH * SEQ / 8, 256, 0, stream>>>(y, gamma, beta);
}